// ComplexMultiHeadedAttention_46093589020880
// MI455X (gfx1250) — compile-verified
//
#include <hip/hip_runtime.h>
#include <hip/hip_bf16.h>
#include <math.h>

// ---------------------------------------------------------------------------
// ComplexMultiHeadedAttention for MI455X (gfx1250), fp32 end-to-end.
// B=4, C=64, T=256, D=256, H=4, DK=64, OUT=256.
// Memory-bound (~1.6 GB HBM traffic vs ~155 GFLOP) -> keep fp32, use
// V_WMMA_F32_16X16X4_F32 for every matmul.
// ---------------------------------------------------------------------------

#define B_  4
#define C_  64
#define T_  256
#define D_  256
#define H_  4
#define DK_ 64
#define NOUT_ 256

#define MIN_VALUE (-3.4028234663852886e+38f)

typedef __attribute__((ext_vector_type(2))) float v2f;
typedef __attribute__((ext_vector_type(8))) float v8f;

// D = A(16x4, f32) x B(4x16, f32) + C(16x16, f32)
__device__ __forceinline__ v8f wmma_f32(v2f a, v2f b, v8f c) {
    return __builtin_amdgcn_wmma_f32_16x16x4_f32(
        /*neg_a=*/false, a, /*neg_b=*/false, b,
        /*c_mod=*/(short)0, c, /*reuse_a=*/false, /*reuse_b=*/false);
}

// ---------------------------------------------------------------------------
// Complex linear:  out_r = (xr@wr + br) - (xi@wi + bi)
//                  out_i = (xi@wr + br) + (xr@wi + bi)
// x: (M=65536, K=256) row-major.  w: (256, 256) row-major.  b: (256,)
// split_heads==1: scatter to (B,H,T,C,DK) layout for attention.
// split_heads==0: out row-major (M, 256), outr/outi separate buffers.
// One wave = one 16x16 output tile; block = 8 waves covering 8 N-tiles.
// grid = (N/128, M/16) = (2, 4096).
// ---------------------------------------------------------------------------
__global__ __launch_bounds__(256) void complex_linear_gemm(
    const float* __restrict__ xr, const float* __restrict__ xi,
    const float* __restrict__ wr, const float* __restrict__ wi,
    const float* __restrict__ br, const float* __restrict__ bi,
    float* __restrict__ outr, float* __restrict__ outi,
    int split_heads)
{
    const int K = 256, N = 256;
    const int lane  = threadIdx.x & 31;
    const int wave  = threadIdx.x >> 5;
    const int col16 = lane & 15;
    const int khalf = (lane >> 4) << 1;   // 0 or 2
    const int rhi   = (lane >> 4) << 3;   // 0 or 8 (C/D row offset)

    const int m0 = blockIdx.y * 16;
    const int n0 = (blockIdx.x * 8 + wave) * 16;

    const float* ar_ptr = xr + (size_t)(m0 + col16) * K + khalf;
    const float* ai_ptr = xi + (size_t)(m0 + col16) * K + khalf;
    const float* br_ptr = wr + (size_t)khalf * N + (n0 + col16);
    const float* bi_ptr = wi + (size_t)khalf * N + (n0 + col16);

    v8f acc_rr = {}; v8f acc_ii = {}; v8f acc_ir = {}; v8f acc_ri = {};

    #pragma unroll 4
    for (int k = 0; k < K; k += 4) {
        v2f a_r = *(const v2f*)(ar_ptr + k);            // global_load_b64
        v2f a_i = *(const v2f*)(ai_ptr + k);
        v2f b_r; b_r.x = br_ptr[(size_t)k * N];  b_r.y = br_ptr[(size_t)(k + 1) * N];
        v2f b_i; b_i.x = bi_ptr[(size_t)k * N];  b_i.y = bi_ptr[(size_t)(k + 1) * N];
        acc_rr = wmma_f32(a_r, b_r, acc_rr);
        acc_ii = wmma_f32(a_i, b_i, acc_ii);
        acc_ir = wmma_f32(a_i, b_r, acc_ir);
        acc_ri = wmma_f32(a_r, b_i, acc_ri);
    }

    const int   n_col = n0 + col16;
    const float brv = br[n_col], biv = bi[n_col];
    const float add_r = brv - biv;
    const float add_i = brv + biv;

    #pragma unroll
    for (int j = 0; j < 8; ++j) {
        const int   m  = m0 + j + rhi;
        const float o_r = acc_rr[j] - acc_ii[j] + add_r;
        const float o_i = acc_ir[j] + acc_ri[j] + add_i;
        if (split_heads) {
            // m = (b*C + c)*T + t ; n_col = h*64 + dk
            const int b = m / (C_ * T_);
            const int c = (m / T_) % C_;
            const int t = m % T_;
            const int h  = n_col >> 6;
            const int dk = n_col & 63;
            const size_t idx = ((((size_t)b * H_ + h) * T_ + t) * C_ + c) * DK_ + dk;
            outr[idx] = o_r;
            outi[idx] = o_i;
        } else {
            const size_t idx = (size_t)m * N + n_col;
            outr[idx] = o_r;
            outi[idx] = o_i;
        }
    }
}

// ---------------------------------------------------------------------------
// Attention core: one block per (b,h,t).  q/k/v in (B,H,T,C,DK) layout.
// S = conj(Q) @ K^T * 1/sqrt(DK); amplitude-softmax over e; X = A @ V.
// X written in (B, C, T, H*DK) layout for the output projection.
// ---------------------------------------------------------------------------
__global__ __launch_bounds__(256) void complex_attention(
    const float* __restrict__ qr, const float* __restrict__ qi,
    const float* __restrict__ kr, const float* __restrict__ ki,
    const float* __restrict__ vr, const float* __restrict__ vi,
    const int*   __restrict__ mask,
    float* __restrict__ xr, float* __restrict__ xi)
{
    __shared__ float s_r[C_][C_];      // 16 KB
    __shared__ float s_i[C_][C_];      // 16 KB
    __shared__ float red[2][C_][4];    // softmax reduction scratch

    const int bht = blockIdx.x;            // (b*H + h)*T + t
    const int b = bht / (H_ * T_);
    const int h = (bht / T_) % H_;
    const int t = bht % T_;

    const size_t base = (size_t)bht * (C_ * DK_);
    const float* Qr = qr + base;  const float* Qi = qi + base;
    const float* Kr = kr + base;  const float* Ki = ki + base;
    const float* Vr = vr + base;  const float* Vi = vi + base;

    const int lane  = threadIdx.x & 31;
    const int wave  = threadIdx.x >> 5;
    const int col16 = lane & 15;
    const int khalf = (lane >> 4) << 1;
    const int rhi   = (lane >> 4) << 3;
    const float scale = 0.125f;  // 1/sqrt(64)

    // ---- Phase A: scores S (64x64) = Q @ K^T (complex, conj(q)) -----------
    #pragma unroll
    for (int tid = wave; tid < 16; tid += 8) {
        const int m0 = (tid >> 2) * 16;
        const int n0 = (tid & 3) * 16;
        v8f rr = {}; v8f ii = {}; v8f ri = {}; v8f ir = {};
        #pragma unroll
        for (int k = 0; k < DK_; k += 4) {
            // A frag: Q rows; B frag: K^T -> contiguous along DK in K rows
            v2f a_r = *(const v2f*)(Qr + (m0 + col16) * DK_ + k + khalf);
            v2f a_i = *(const v2f*)(Qi + (m0 + col16) * DK_ + k + khalf);
            v2f b_r = *(const v2f*)(Kr + (n0 + col16) * DK_ + k + khalf);
            v2f b_i = *(const v2f*)(Ki + (n0 + col16) * DK_ + k + khalf);
            rr = wmma_f32(a_r, b_r, rr);
            ii = wmma_f32(a_i, b_i, ii);
            ri = wmma_f32(a_r, b_i, ri);
            ir = wmma_f32(a_i, b_r, ir);
        }
        #pragma unroll
        for (int j = 0; j < 8; ++j) {
            s_r[m0 + j + rhi][n0 + col16] = (rr[j] + ii[j]) * scale;
            s_i[m0 + j + rhi][n0 + col16] = (ri[j] - ir[j]) * scale;
        }
    }
    __syncthreads();

    // ---- Amplitude softmax over e (row-wise), with mask -------------------
    {
        const int row = threadIdx.x >> 2;   // 0..63
        const int q4  = threadIdx.x & 3;
        const int c0  = q4 * 16;
        float amp_l[16];
        float lmax = MIN_VALUE;
        #pragma unroll
        for (int j = 0; j < 16; ++j) {
            const int e = c0 + j;
            const float sr = s_r[row][e];
            const float si = s_i[row][e];
            float a = sqrtf(sr * sr + si * si);
            if (mask[b * C_ + e] == 0) a = MIN_VALUE;
            amp_l[j] = a;
            lmax = fmaxf(lmax, a);
        }
        red[0][row][q4] = lmax;
        __syncthreads();
        const float rmax = fmaxf(fmaxf(red[0][row][0], red[0][row][1]),
                                 fmaxf(red[0][row][2], red[0][row][3]));
        float lsum = 0.f;
        #pragma unroll
        for (int j = 0; j < 16; ++j) lsum += expf(amp_l[j] - rmax);
        red[1][row][q4] = lsum;
        __syncthreads();
        const float inv = 1.0f / (red[1][row][0] + red[1][row][1] +
                                  red[1][row][2] + red[1][row][3]);
        #pragma unroll
        for (int j = 0; j < 16; ++j) {
            const int e = c0 + j;
            const bool msk = (mask[b * C_ + e] == 0);
            const float sc = expf(amp_l[j] - rmax) * inv / amp_l[j];
            const float sr = s_r[row][e];
            const float si = s_i[row][e];
            s_r[row][e] = msk ? 0.f : sc * sr;
            s_i[row][e] = msk ? 0.f : sc * si;
        }
    }
    __syncthreads();

    // ---- Phase B: X (64x64) = A @ V (complex) -----------------------------
    #pragma unroll
    for (int tid = wave; tid < 16; tid += 8) {
        const int m0 = (tid >> 2) * 16;   // c tile
        const int n0 = (tid & 3) * 16;    // d tile
        v8f arvr = {}; v8f aivi = {}; v8f aivr = {}; v8f arvi = {};
        #pragma unroll
        for (int k = 0; k < C_; k += 4) {
            // A frag from LDS (contiguous -> ds_load_b64)
            v2f a_r = *(const v2f*)(&s_r[m0 + col16][k + khalf]);
            v2f a_i = *(const v2f*)(&s_i[m0 + col16][k + khalf]);
            v2f b_r; b_r.x = Vr[(k + khalf) * DK_ + n0 + col16];
                     b_r.y = Vr[(k + khalf + 1) * DK_ + n0 + col16];
            v2f b_i; b_i.x = Vi[(k + khalf) * DK_ + n0 + col16];
                     b_i.y = Vi[(k + khalf + 1) * DK_ + n0 + col16];
            arvr = wmma_f32(a_r, b_r, arvr);
            aivi = wmma_f32(a_i, b_i, aivi);
            aivr = wmma_f32(a_i, b_r, aivr);
            arvi = wmma_f32(a_r, b_i, arvi);
        }
        #pragma unroll
        for (int j = 0; j < 8; ++j) {
            const int c = m0 + j + rhi;
            const int d = n0 + col16;
            const size_t idx = (((size_t)b * C_ + c) * T_ + t) * (H_ * DK_) + h * DK_ + d;
            xr[idx] = arvr[j] - aivi[j];
            xi[idx] = aivr[j] + arvi[j];
        }
    }
}

// ---------------------------------------------------------------------------
extern "C" void kernel_launch(void* const* d_in, const int* in_sizes, int n_in,
                              void* d_out, int out_size, void* d_ws, size_t ws_size,
                              hipStream_t stream) {
    const float* q_r  = (const float*)d_in[0];
    const float* q_i  = (const float*)d_in[1];
    const float* k_r  = (const float*)d_in[2];
    const float* k_i  = (const float*)d_in[3];
    const float* v_r  = (const float*)d_in[4];
    const float* v_i  = (const float*)d_in[5];
    const int*   mask = (const int*)  d_in[6];
    const float* wq_r = (const float*)d_in[7];
    const float* wq_i = (const float*)d_in[8];
    const float* bq_r = (const float*)d_in[9];
    const float* bq_i = (const float*)d_in[10];
    const float* wk_r = (const float*)d_in[11];
    const float* wk_i = (const float*)d_in[12];
    const float* bk_r = (const float*)d_in[13];
    const float* bk_i = (const float*)d_in[14];
    const float* wv_r = (const float*)d_in[15];
    const float* wv_i = (const float*)d_in[16];
    const float* bv_r = (const float*)d_in[17];
    const float* bv_i = (const float*)d_in[18];
    const float* wo_r = (const float*)d_in[19];
    const float* wo_i = (const float*)d_in[20];
    const float* bo_r = (const float*)d_in[21];
    const float* bo_i = (const float*)d_in[22];

    // Workspace: 8 fp32 arrays of B*H*T*C*DK = 16,777,216 elements (64 MB each)
    const size_t NE = (size_t)B_ * H_ * T_ * C_ * DK_;
    float* ws   = (float*)d_ws;
    float* qr_p = ws + 0 * NE;  float* qi_p = ws + 1 * NE;
    float* kr_p = ws + 2 * NE;  float* ki_p = ws + 3 * NE;
    float* vr_p = ws + 4 * NE;  float* vi_p = ws + 5 * NE;
    float* xr_p = ws + 6 * NE;  float* xi_p = ws + 7 * NE;

    const dim3 blk(256);
    const dim3 gGemm(NOUT_ / 128, (B_ * C_ * T_) / 16);  // (2, 4096)

    // Stage 1: complex q/k/v projections, scattered straight into head layout
    complex_linear_gemm<<<gGemm, blk, 0, stream>>>(q_r, q_i, wq_r, wq_i, bq_r, bq_i, qr_p, qi_p, 1);
    complex_linear_gemm<<<gGemm, blk, 0, stream>>>(k_r, k_i, wk_r, wk_i, bk_r, bk_i, kr_p, ki_p, 1);
    complex_linear_gemm<<<gGemm, blk, 0, stream>>>(v_r, v_i, wv_r, wv_i, bv_r, bv_i, vr_p, vi_p, 1);

    // Stage 2: attention per (b,h,t)
    complex_attention<<<dim3(B_ * H_ * T_), blk, 0, stream>>>(
        qr_p, qi_p, kr_p, ki_p, vr_p, vi_p, mask, xr_p, xi_p);

    // Stage 3: output projection -> d_out = [out_r | out_i]
    float* out_r = (float*)d_out;
    float* out_i = out_r + (size_t)B_ * C_ * T_ * NOUT_;
    complex_linear_gemm<<<gGemm, blk, 0, stream>>>(xr_p, xi_p, wo_r, wo_i, bo_r, bo_i, out_r, out_i, 0);
}